// ResNetCAMCONRoIHead_63385127354457
// MI455X (gfx1250) — compile-verified
//
#include <hip/hip_runtime.h>
#include <hip/hip_bf16.h>
#include <math.h>

typedef __attribute__((ext_vector_type(2))) float v2f;
typedef __attribute__((ext_vector_type(8))) float v8f;

#define C_SLOW 2048
#define C_FAST 256
#define C_TOT  2304
#define T_SLOW 8
#define T_FAST 32
#define HDIM   14
#define WDIM   14
#define HW     196
#define NBINS  49      // 7x7 ROIAlign bins
#define NPAD   64      // bins padded to 4 WMMA N-tiles
#define NROI   256     // K*NB = 32*8
#define DOUT   512
#define KC2    28      // kernel-2 A-chunk along K (7 wmma k-steps)
#define BSTR   36      // kernel-3 LDS W-tile row stride (16B-aligned, bank-spread)
#define NEG_BIG (-3.402823466e38f)

// Pair layout for kernel-2 M matrix: Mp[k>>1][n] = {M[k][n], M[k+1][n]} so a
// WMMA B fragment (k even per lane-half) is one aligned 8-byte LDS load.
#define MIDX(hw, bin) ((((hw) >> 1) * NPAD + (bin)) * 2 + ((hw) & 1))

// ---------------------------------------------------------------------------
// CDNA5 async global->LDS copies (ASYNCcnt-tracked) via inline asm; the clang
// builtins don't expose these portably.  Each active lane copies 4/16 bytes
// from its 64-bit global address to (workgroup LDS base + lds_off).
// ---------------------------------------------------------------------------
__device__ __forceinline__ void async_copy_b128(unsigned lds_off, const float* g) {
  asm volatile("global_load_async_to_lds_b128 %0, %1, off"
               :: "v"(lds_off), "v"((unsigned long long)(uintptr_t)g)
               : "memory");
}
__device__ __forceinline__ void async_wait0() {
  asm volatile("s_wait_asynccnt 0" ::: "memory");
}

// ---------------------------------------------------------------------------
// Kernel 1: temporal mean (AvgPool3d over T) into fused fm[b][c][hw], c<2048
// from slow (T=8), c>=2048 from fast (T=32).  HBM-bound: 154 MB streamed once
// -> ~6.6us floor at 23.3 TB/s.  float4 vectorized (49 float4 per 196-row).
// ---------------------------------------------------------------------------
__global__ __launch_bounds__(256) void temporal_mean_kernel(
    const float* __restrict__ slow, const float* __restrict__ fast,
    float* __restrict__ fm) {
  int idx = blockIdx.x * blockDim.x + threadIdx.x;
  const int total = 8 * C_TOT * (HW / 4);
  if (idx >= total) return;
  int q = idx % (HW / 4);
  int c = (idx / (HW / 4)) % C_TOT;
  int b = idx / ((HW / 4) * C_TOT);
  float4 acc = make_float4(0.f, 0.f, 0.f, 0.f);
  if (c < C_SLOW) {
    const float4* p =
        (const float4*)(slow + ((size_t)b * C_SLOW + c) * T_SLOW * HW) + q;
    __builtin_prefetch(p + T_SLOW * (HW / 4), 0, 1);  // global_prefetch_b8
#pragma unroll
    for (int t = 0; t < T_SLOW; ++t) {
      float4 v = p[t * (HW / 4)];
      acc.x += v.x; acc.y += v.y; acc.z += v.z; acc.w += v.w;
    }
    const float s = 1.0f / (float)T_SLOW;
    acc.x *= s; acc.y *= s; acc.z *= s; acc.w *= s;
  } else {
    const float4* p =
        (const float4*)(fast + ((size_t)b * C_FAST + (c - C_SLOW)) * T_FAST * HW) + q;
    __builtin_prefetch(p + T_FAST * (HW / 4), 0, 1);
#pragma unroll
    for (int t = 0; t < T_FAST; ++t) {
      float4 v = p[t * (HW / 4)];
      acc.x += v.x; acc.y += v.y; acc.z += v.z; acc.w += v.w;
    }
    const float s = 1.0f / (float)T_FAST;
    acc.x *= s; acc.y *= s; acc.z *= s; acc.w *= s;
  }
  ((float4*)(fm + ((size_t)b * C_TOT + c) * HW))[q] = acc;
}

// ---------------------------------------------------------------------------
// Kernel 2: per-ROI ROIAlign-as-GEMM + 7x7 max pool.
// pooled[c,bin] = sum_hw fm[b][c][hw] * M[hw][bin]; M built in LDS (pair
// layout) from the bilinear tap weights (masks + 1/count folded, ds_add_f32).
// GEMM: [2304 x 196] @ [196 x 64pad] via V_WMMA_F32_16X16X4_F32, 8 waves.
// A tiles staged per-wave into LDS with async b128 global->LDS copies, B/A
// fragments are single conflict-free ds_load_b64s.  One workgroup per ROI.
// LDS: 196*64*4 (M pairs) + 8 * 16*28*4 (A slabs) = 64512 B.
// ---------------------------------------------------------------------------
__global__ __launch_bounds__(256) void roialign_wmma_kernel(
    const float* __restrict__ fm, const float* __restrict__ rois,
    float* __restrict__ X) {
  __shared__ float pool[HW * NPAD + 8 * 16 * KC2];
  float* Msh = pool;                       // LDS byte offset 0
  const int roi = blockIdx.x;
  const int tid = threadIdx.x;

  for (int i = tid; i < HW * NPAD; i += 256) Msh[i] = 0.0f;
  __syncthreads();

  const float* r = rois + roi * 5;
  const int bidx = (int)r[0];
  const float SC = 1.0f / 16.0f;
  const float x1 = r[1] * SC - 0.5f, y1 = r[2] * SC - 0.5f;
  const float x2 = r[3] * SC - 0.5f, y2 = r[4] * SC - 0.5f;
  const float roiw = x2 - x1, roih = y2 - y1;
  const float binw = roiw / 7.0f, binh = roih / 7.0f;
  const int gh = (int)fminf(fmaxf(ceilf(roih / 7.0f), 1.0f), 2.0f);
  const int gw = (int)fminf(fmaxf(ceilf(roiw / 7.0f), 1.0f), 2.0f);
  const float inv_count = 1.0f / (float)(gh * gw);

  if (tid < NBINS * 4) {
    const int bin = tid >> 2;
    const int smp = tid & 3;
    const int p = bin / 7, q = bin % 7;
    const int i = smp >> 1, j = smp & 1;
    if (i < gh && j < gw) {
      const float ys = y1 + ((float)p + ((float)i + 0.5f) / (float)gh) * binh;
      const float xs = x1 + ((float)q + ((float)j + 0.5f) / (float)gw) * binw;
      const bool vy = (ys > -1.0f) && (ys < (float)HDIM);
      const bool vx = (xs > -1.0f) && (xs < (float)WDIM);
      if (vy && vx) {
        const float y = fminf(fmaxf(ys, 0.0f), (float)(HDIM - 1));
        const float x = fminf(fmaxf(xs, 0.0f), (float)(WDIM - 1));
        const int y0 = (int)floorf(y), x0 = (int)floorf(x);
        const int y1i = (y0 + 1 > HDIM - 1) ? HDIM - 1 : y0 + 1;
        const int x1i = (x0 + 1 > WDIM - 1) ? WDIM - 1 : x0 + 1;
        const float ly = y - (float)y0, lx = x - (float)x0;
        const float wyl = (1.0f - ly) * inv_count, wyh = ly * inv_count;
        const float wxl = 1.0f - lx, wxh = lx;
        atomicAdd(&Msh[MIDX(y0 * WDIM + x0, bin)], wyl * wxl);
        atomicAdd(&Msh[MIDX(y0 * WDIM + x1i, bin)], wyl * wxh);
        atomicAdd(&Msh[MIDX(y1i * WDIM + x0, bin)], wyh * wxl);
        atomicAdd(&Msh[MIDX(y1i * WDIM + x1i, bin)], wyh * wxh);
      }
    }
  }
  __syncthreads();

  const int wave = tid >> 5;            // 8 waves
  const int lane = tid & 31;
  const int lrow = lane & 15;
  const int koff = (lane < 16) ? 0 : 2; // K split across lane halves
  const float* A = fm + (size_t)bidx * C_TOT * HW;

  // per-wave A slab: 16 rows x KC2 cols, single-buffered (reads are waited
  // into VGPRs before the next chunk's async copies issue).
  float* AshW = pool + HW * NPAD + wave * 16 * KC2;
  const unsigned ashOff = (unsigned)((HW * NPAD + wave * 16 * KC2) * 4);

  for (int mt = wave; mt < C_TOT / 16; mt += 8) {  // 144/8 = 18, uniform
    const float* Abase = A + (size_t)(mt * 16) * HW;
    v8f acc[4];
#pragma unroll
    for (int nt = 0; nt < 4; ++nt) acc[nt] = (v8f){0.f,0.f,0.f,0.f,0.f,0.f,0.f,0.f};

    for (int k0 = 0; k0 < HW; k0 += KC2) {  // 7 chunks of 28
      // async-stage A[16][KC2] as 112 b128 copies (16B-aligned both sides)
      for (int e = lane; e < 16 * (KC2 / 4); e += 32) {
        const int row = e / (KC2 / 4);
        const int c4 = e - row * (KC2 / 4);
        async_copy_b128(ashOff + (unsigned)(row * KC2 + c4 * 4) * 4u,
                        Abase + (size_t)row * HW + k0 + c4 * 4);
      }
      async_wait0();
#pragma unroll
      for (int kk = 0; kk < KC2; kk += 4) {
        v2f a = *(const v2f*)&AshW[lrow * KC2 + kk + koff];
        const float* Bp = &Msh[(unsigned)((k0 + kk + koff) >> 1) * (NPAD * 2)];
#pragma unroll
        for (int nt = 0; nt < 4; ++nt) {
          v2f bfrag = *(const v2f*)&Bp[(nt * 16 + lrow) * 2];
          acc[nt] = __builtin_amdgcn_wmma_f32_16x16x4_f32(
              false, a, false, bfrag, (short)0, acc[nt], false, false);
        }
      }
    }

    // masked max over the 49 valid bins, then 16-lane shfl reduction per row
#pragma unroll
    for (int rx = 0; rx < 8; ++rx) {
      float m = acc[0][rx];                       // cols 0..15 valid
      m = fmaxf(m, acc[1][rx]);                   // cols 16..31 valid
      m = fmaxf(m, acc[2][rx]);                   // cols 32..47 valid
      float m3 = (48 + lrow < NBINS) ? acc[3][rx] : NEG_BIG;  // only col 48
      m = fmaxf(m, m3);
#pragma unroll
      for (int off = 8; off; off >>= 1) m = fmaxf(m, __shfl_xor(m, off, 32));
      if (lrow == 0) {
        const int row = mt * 16 + ((lane < 16) ? 0 : 8) + rx;
        X[(size_t)roi * C_TOT + row] = m;
      }
    }
  }
}

// ---------------------------------------------------------------------------
// Kernel 3: projection  out[256,512] = X[256,2304] @ W^T + b  via fp32 WMMA.
// One wave per block; block (mt, ngroup-of-4-Ntiles). W tile staged in LDS
// [n][k] with row stride 36 (16B-aligned for b128 async copies; n*36 mod 64
// distinct over 16 lanes -> conflict-free b64 fragment reads).
// ---------------------------------------------------------------------------
__global__ __launch_bounds__(32) void proj_wmma_kernel(
    const float* __restrict__ X, const float* __restrict__ Wm,
    const float* __restrict__ bias, float* __restrict__ out) {
  __shared__ float Bs[64 * BSTR];  // 9216 B, LDS byte offset 0
  const int mt = blockIdx.x;       // 16 M-tiles
  const int ng = blockIdx.y;       // 8 groups of 64 cols
  const int lane = threadIdx.x;
  const int lrow = lane & 15;
  const int koff = (lane < 16) ? 0 : 2;
  const int nbase = ng * 64;

  v8f acc[4];
#pragma unroll
  for (int nt = 0; nt < 4; ++nt) acc[nt] = (v8f){0.f,0.f,0.f,0.f,0.f,0.f,0.f,0.f};

  const float* Arow = X + (size_t)(mt * 16 + lrow) * C_TOT;

  for (int k0 = 0; k0 < C_TOT; k0 += 32) {
    // async-stage W[nbase..nbase+63][k0..k0+31] as Bs[n][kk], b128 copies
    for (int e = lane; e < 64 * 8; e += 32) {   // 512 float4, 16/lane
      const int n = e >> 3;
      const int k4 = e & 7;
      async_copy_b128((unsigned)((n * BSTR + k4 * 4) * 4),
                      Wm + (size_t)(nbase + n) * C_TOT + k0 + k4 * 4);
    }
    async_wait0();   // single wave per block: no barrier needed
#pragma unroll
    for (int kk = 0; kk < 32; kk += 4) {
      const int k = k0 + kk;
      v2f a;
      a.x = Arow[k + koff];
      a.y = Arow[k + koff + 1];
#pragma unroll
      for (int nt = 0; nt < 4; ++nt) {
        v2f bfrag = *(const v2f*)&Bs[(nt * 16 + lrow) * BSTR + kk + koff];
        acc[nt] = __builtin_amdgcn_wmma_f32_16x16x4_f32(
            false, a, false, bfrag, (short)0, acc[nt], false, false);
      }
    }
  }

#pragma unroll
  for (int nt = 0; nt < 4; ++nt) {
    const int col = nbase + nt * 16 + lrow;
    const float bv = bias[col];
#pragma unroll
    for (int rx = 0; rx < 8; ++rx) {
      const int row = mt * 16 + ((lane < 16) ? 0 : 8) + rx;
      out[(size_t)row * DOUT + col] = acc[nt][rx] + bv;
    }
  }
}

extern "C" void kernel_launch(void* const* d_in, const int* in_sizes, int n_in,
                              void* d_out, int out_size, void* d_ws, size_t ws_size,
                              hipStream_t stream) {
  const float* slow = (const float*)d_in[0];  // [8,2048,8,14,14]
  const float* fast = (const float*)d_in[1];  // [8,256,32,14,14]
  const float* rois = (const float*)d_in[2];  // [32,8,5] -> 256 x 5
  const float* Wm   = (const float*)d_in[3];  // [512,2304]
  const float* bias = (const float*)d_in[4];  // [512]
  float* out = (float*)d_out;                 // [32,8,512] flat

  float* fm = (float*)d_ws;                        // [8][2304][196]
  float* X  = fm + (size_t)8 * C_TOT * HW;         // [256][2304]

  {
    const int total = 8 * C_TOT * (HW / 4);
    temporal_mean_kernel<<<(total + 255) / 256, 256, 0, stream>>>(slow, fast, fm);
  }
  roialign_wmma_kernel<<<NROI, 256, 0, stream>>>(fm, rois, X);
  {
    dim3 grid(256 / 16, DOUT / 64);  // 16 x 8
    proj_wmma_kernel<<<grid, 32, 0, stream>>>(X, Wm, bias, out);
  }
}